// FeaStEncoderBlock_5849745457495
// MI455X (gfx1250) — compile-verified
//
#include <hip/hip_runtime.h>

#define NHEADS 4

typedef float v2f __attribute__((ext_vector_type(2)));
typedef float v8f __attribute__((ext_vector_type(8)));

// ---------------------------------------------------------------------------
// C[M x Cout] = A[M x K] * W[K x Cout], fp32 WMMA 16x16x4.
// One wave computes a 16(M) x 64(N) tile with 4 v8f accumulators.
// Requires: M % 16 == 0, K % 4 == 0, Cout % 64 == 0 (all exact for this net).
// permute!=0 (Cout==1024): output column col=(h*256+ch) stored at
// row*1024 + ch*4 + h  so the edge kernel reads 4 heads of a channel as float4.
// ---------------------------------------------------------------------------
__global__ __launch_bounds__(256)
void wmma_gemm_f32(const float* __restrict__ A, const float* __restrict__ W,
                   float* __restrict__ C, int M, int K, int Cout, int permute)
{
    const int lane       = threadIdx.x & 31;
    const int waveInBlk  = threadIdx.x >> 5;
    const int waveId     = blockIdx.x * (blockDim.x >> 5) + waveInBlk;
    const int nColGroups = Cout >> 6;                 // 64-wide column groups
    const int mTile      = waveId / nColGroups;
    const int cg         = waveId % nColGroups;
    if (mTile * 16 >= M) return;                      // wave-uniform exit

    const int half = lane >> 4;                       // 0: lanes 0-15, 1: 16-31
    const int sub  = lane & 15;
    const int row  = mTile * 16 + sub;                // A row this lane feeds
    const int n0   = cg * 64;

    v8f acc0 = {}, acc1 = {}, acc2 = {}, acc3 = {};

    const float* __restrict__ arow = A + (size_t)row * K;
    for (int k0 = 0; k0 < K; k0 += 4) {
        // A 16x4: lanes 0-15 hold K = k0+0,k0+1 ; lanes 16-31 hold K = k0+2,k0+3
        v2f a;
        a.x = arow[k0 + 2 * half + 0];
        a.y = arow[k0 + 2 * half + 1];
        // B 4x16 per tile: VGPR0 = row (k0+2*half), VGPR1 = row (k0+2*half+1)
        const float* __restrict__ wb = W + (size_t)(k0 + 2 * half) * Cout;
        const int ci = n0 + sub;
        v2f b0, b1, b2, b3;
        b0.x = wb[ci +  0]; b0.y = wb[Cout + ci +  0];
        b1.x = wb[ci + 16]; b1.y = wb[Cout + ci + 16];
        b2.x = wb[ci + 32]; b2.y = wb[Cout + ci + 32];
        b3.x = wb[ci + 48]; b3.y = wb[Cout + ci + 48];
        acc0 = __builtin_amdgcn_wmma_f32_16x16x4_f32(false, a, false, b0, (short)0, acc0, false, false);
        acc1 = __builtin_amdgcn_wmma_f32_16x16x4_f32(false, a, false, b1, (short)0, acc1, false, false);
        acc2 = __builtin_amdgcn_wmma_f32_16x16x4_f32(false, a, false, b2, (short)0, acc2, false, false);
        acc3 = __builtin_amdgcn_wmma_f32_16x16x4_f32(false, a, false, b3, (short)0, acc3, false, false);
    }

    // C/D layout: VGPR r -> M = r (lanes 0-15) / r+8 (lanes 16-31), N = sub
    #pragma unroll
    for (int t = 0; t < 4; ++t) {
        v8f acc = (t == 0) ? acc0 : (t == 1) ? acc1 : (t == 2) ? acc2 : acc3;
        const int col = n0 + t * 16 + sub;
        #pragma unroll
        for (int r = 0; r < 8; ++r) {
            const int orow = mTile * 16 + r + 8 * half;
            const float v = acc[r];
            if (permute) {
                const int h = col >> 8, ch = col & 255;
                C[(size_t)orow * 1024 + ch * 4 + h] = v;
            } else {
                C[(size_t)orow * Cout + col] = v;
            }
        }
    }
}

// ---------------------------------------------------------------------------
// One wave per edge (incl. synthesized self-loops e >= E).
// y is permuted [N][256][4] (channel-major, heads innermost).
// ---------------------------------------------------------------------------
template <int IN_CH>
__global__ __launch_bounds__(256)
void feast_edge(const float* __restrict__ x, const float* __restrict__ y,
                const int* __restrict__ ei,                 // [2][E]
                const float* __restrict__ u,                // [IN_CH][4]
                const float* __restrict__ cvec,             // [4]
                float* __restrict__ s, float* __restrict__ cnt,
                int E, int totalE, int addCnt)
{
    const int lane = threadIdx.x & 31;
    const int wave = (int)((blockIdx.x * blockDim.x + threadIdx.x) >> 5);
    if (wave >= totalE) return;                             // wave-uniform

    int srcN, dstN;
    if (wave < E) { srcN = ei[wave]; dstN = ei[E + wave]; }
    else          { srcN = dstN = wave - E; }

    const float* __restrict__ xs = x + (size_t)srcN * IN_CH;
    const float* __restrict__ xd = x + (size_t)dstN * IN_CH;
    const float* __restrict__ ys = y + (size_t)srcN * 1024;
    __builtin_prefetch(ys + lane * 32, 0, 1);               // pull y row toward L2/L0

    // head logits: acc[h] = sum_ch (x_s - x_d)[ch] * u[ch][h]
    float acc[NHEADS] = {0.f, 0.f, 0.f, 0.f};
    #pragma unroll
    for (int i = 0; i < IN_CH / 32; ++i) {
        const int ch = lane + 32 * i;
        const float d = xs[ch] - xd[ch];
        const float4 uu = *(const float4*)(u + ch * 4);
        acc[0] += d * uu.x; acc[1] += d * uu.y;
        acc[2] += d * uu.z; acc[3] += d * uu.w;
    }
    #pragma unroll
    for (int h = 0; h < NHEADS; ++h) {
        #pragma unroll
        for (int off = 16; off; off >>= 1)
            acc[h] += __shfl_xor(acc[h], off, 32);
        acc[h] += cvec[h];
    }
    // softmax over 4 heads (wave-uniform values)
    float mx = fmaxf(fmaxf(acc[0], acc[1]), fmaxf(acc[2], acc[3]));
    float e0 = __expf(acc[0] - mx), e1 = __expf(acc[1] - mx);
    float e2 = __expf(acc[2] - mx), e3 = __expf(acc[3] - mx);
    const float inv = 1.0f / (e0 + e1 + e2 + e3);
    const float q0 = e0 * inv, q1 = e1 * inv, q2 = e2 * inv, q3 = e3 * inv;

    // msg[ch] = sum_h q_h * y[src][ch][h]; scatter-add into s[dst]
    float* __restrict__ sd = s + (size_t)dstN * 256;
    #pragma unroll
    for (int j = 0; j < 8; ++j) {
        const int ch = lane + 32 * j;
        const float4 yv = *(const float4*)(ys + ch * 4);
        const float m = q0 * yv.x + q1 * yv.y + q2 * yv.z + q3 * yv.w;
        atomicAdd(sd + ch, m);
    }
    if (addCnt && lane == 0) atomicAdd(cnt + dstN, 1.0f);
}

// ---------------------------------------------------------------------------
// out = relu(s / max(cnt,1) + bias + skip (+ skipb))
// ---------------------------------------------------------------------------
__global__ __launch_bounds__(256)
void finalize_kernel(const float* __restrict__ s, const float* __restrict__ cnt,
                     const float* __restrict__ bias, const float* __restrict__ skip,
                     const float* __restrict__ skipb, float* __restrict__ out, int total)
{
    const int i = blockIdx.x * blockDim.x + threadIdx.x;
    if (i >= total) return;
    const int n = i >> 8, ch = i & 255;
    float c = cnt[n]; c = (c < 1.0f) ? 1.0f : c;
    float v = s[i] / c + bias[ch] + skip[i];
    if (skipb) v += skipb[ch];
    out[i] = (v > 0.0f) ? v : 0.0f;
}

__global__ __launch_bounds__(256)
void copy_words(const int* __restrict__ in, int* __restrict__ out, int n)
{
    const int i = blockIdx.x * blockDim.x + threadIdx.x;
    if (i < n) out[i] = in[i];
}

// ---------------------------------------------------------------------------
extern "C" void kernel_launch(void* const* d_in, const int* in_sizes, int n_in,
                              void* d_out, int out_size, void* d_ws, size_t ws_size,
                              hipStream_t stream)
{
    const float* x      = (const float*)d_in[0];
    const int*   ei     = (const int*)  d_in[1];
    const float* W0     = (const float*)d_in[2];
    const float* u0     = (const float*)d_in[3];
    const float* c0     = (const float*)d_in[4];
    const float* b0     = (const float*)d_in[5];
    const float* skipW0 = (const float*)d_in[6];
    const float* skipb0 = (const float*)d_in[7];
    const float* W1     = (const float*)d_in[8];
    const float* u1     = (const float*)d_in[9];
    const float* c1     = (const float*)d_in[10];
    const float* b1     = (const float*)d_in[11];

    const int N      = in_sizes[0] / 128;
    const int E      = in_sizes[1] / 2;
    const int totalE = E + N;

    // workspace carve-up (floats, 256B-aligned chunks)
    float* ws  = (float*)d_ws;
    size_t off = 0;
    auto carve = [&](size_t ne) { float* p = ws + off; off += (ne + 63) & ~(size_t)63; return p; };
    float* y    = carve((size_t)N * 1024);   // permuted head projections (reused blk0/blk1)
    float* sbuf = carve((size_t)N * 256);    // segment sums (reused)
    float* cnt  = carve((size_t)N);          // in-degrees (shared by both blocks)
    float* skip = carve((size_t)N * 256);    // x @ skipW0
    float* h    = carve((size_t)N * 256);    // block-0 output

    const int edgeBlocks = (totalE + 7) / 8;       // 8 waves / 256-thread block
    const int finThreads = N * 256;

    // ---- Block 0: FeaStConv(128->256) + linear skip ----
    {
        const int waves = (N / 16) * (1024 / 64);
        wmma_gemm_f32<<<(waves + 7) / 8, 256, 0, stream>>>(x, W0, y, N, 128, 1024, 1);
    }
    {
        const int waves = (N / 16) * (256 / 64);
        wmma_gemm_f32<<<(waves + 7) / 8, 256, 0, stream>>>(x, skipW0, skip, N, 128, 256, 0);
    }
    hipMemsetAsync(sbuf, 0, (size_t)N * 256 * sizeof(float), stream);
    hipMemsetAsync(cnt,  0, (size_t)N * sizeof(float), stream);
    feast_edge<128><<<edgeBlocks, 256, 0, stream>>>(x, y, ei, u0, c0, sbuf, cnt, E, totalE, 1);
    finalize_kernel<<<(finThreads + 255) / 256, 256, 0, stream>>>(sbuf, cnt, b0, skip, skipb0, h, finThreads);

    // ---- Block 1: FeaStConv(256->256) + identity skip ----
    {
        const int waves = (N / 16) * (1024 / 64);
        wmma_gemm_f32<<<(waves + 7) / 8, 256, 0, stream>>>(h, W1, y, N, 256, 1024, 1);
    }
    hipMemsetAsync(sbuf, 0, (size_t)N * 256 * sizeof(float), stream);
    feast_edge<256><<<edgeBlocks, 256, 0, stream>>>(h, y, ei, u1, c1, sbuf, cnt, E, totalE, 0);
    finalize_kernel<<<(finThreads + 255) / 256, 256, 0, stream>>>(sbuf, cnt, b1, h, nullptr,
                                                                  (float*)d_out, finThreads);

    // ---- pass-through edge_index into tail of d_out ----
    const int tail = out_size - N * 256;
    if (tail > 0) {
        copy_words<<<(tail + 255) / 256, 256, 0, stream>>>(ei, (int*)d_out + (size_t)N * 256, tail);
    }
}